// Decoder_20375324852253
// MI455X (gfx1250) — compile-verified
//
#include <hip/hip_runtime.h>
#include <math.h>

// Problem constants (from the reference)
#define BATCH 64
#define NNODE 512
#define DIM   128
#define NHEAD 8
#define DK    16
#define SCALE 0.25f          // 1/sqrt(16)
#define NWAVE 16             // 512 threads / wave32

typedef __attribute__((ext_vector_type(2))) float v2f;
typedef __attribute__((ext_vector_type(8))) float v8f;
typedef unsigned int u32x4 __attribute__((ext_vector_type(4)));
typedef int          i32x4 __attribute__((ext_vector_type(4)));
typedef int          i32x8 __attribute__((ext_vector_type(8)));

#if defined(__has_builtin)
#if __has_builtin(__builtin_amdgcn_tensor_load_to_lds) && __has_builtin(__builtin_amdgcn_s_wait_tensorcnt)
#define USE_TDM 1
#endif
#endif
#ifndef USE_TDM
#define USE_TDM 0
#endif

// ---------------------------------------------------------------------------
// Kernel A: precompute (verified WMMA codegen in round 1)
//   Kall[b][n][h*16+k] = sum_d h[b][n][d] * W_K[h][d][k]      (f32 WMMA GEMM)
//   qg[b][h*16+k]      = sum_d h_graph[b][d] * W_Q_graph[h][d][k]
//   Wqn_t[j][d]        = W_Q_nodes[j>>4][d][j&15]             (transpose once)
// ---------------------------------------------------------------------------
__global__ __launch_bounds__(256)
void precompute_kernel(const float* __restrict__ h,
                       const float* __restrict__ Wk,
                       const float* __restrict__ hg,
                       const float* __restrict__ Wqg,
                       const float* __restrict__ Wqn,
                       float* __restrict__ Kall,
                       float* __restrict__ qg,
                       float* __restrict__ Wqn_t) {
    const int b    = blockIdx.x;
    const int lane = threadIdx.x & 31;
    const int wave = threadIdx.x >> 5;          // 0..7
    const int m    = lane & 15;                 // row-in-tile (A) / col (B,C)
    const int kb2  = (lane >> 4) << 1;          // K base: lanes 0-15 -> {0,1}, 16-31 -> {2,3}

    const float* A = h + (size_t)b * NNODE * DIM;

    for (int rt = wave; rt < NNODE / 16; rt += 8) {
        const int row0 = rt * 16;
        for (int c = 0; c < NHEAD; ++c) {
            v8f acc = {};
            const float* Bw = Wk + c * (DIM * DK);   // head c: [128][16]
            for (int kk = 0; kk < DIM; kk += 4) {
                v2f av, bv;
                av.x = A[(size_t)(row0 + m) * DIM + kk + kb2];
                av.y = A[(size_t)(row0 + m) * DIM + kk + kb2 + 1];
                bv.x = Bw[(kk + kb2) * DK + m];
                bv.y = Bw[(kk + kb2 + 1) * DK + m];
                acc = __builtin_amdgcn_wmma_f32_16x16x4_f32(
                        false, av, false, bv, (short)0, acc, false, false);
            }
            const int rbase = (lane >> 4) << 3;
            for (int r = 0; r < 8; ++r) {
                Kall[((size_t)b * NNODE + row0 + rbase + r) * DIM + c * 16 + m] = acc[r];
            }
        }
    }

    if (threadIdx.x < DIM) {
        const int j  = threadIdx.x;
        const int hh = j >> 4, k = j & 15;
        const float* hgb = hg + (size_t)b * DIM;
        const float* w   = Wqg + hh * (DIM * DK) + k;
        float s = 0.f;
        for (int d = 0; d < DIM; ++d) s += hgb[d] * w[d * DK];
        qg[(size_t)b * DIM + j] = s;
    }

    if (b == 0) {
        for (int idx = threadIdx.x; idx < DIM * 2 * DIM; idx += blockDim.x) {
            const int j = idx >> 8;          // 0..127  (head*16 + dk)
            const int d = idx & 255;         // 0..255
            Wqn_t[idx] = Wqn[(size_t)(j >> 4) * (2 * DIM * DK) + d * DK + (j & 15)];
        }
    }
}

// ---------------------------------------------------------------------------
// Wave32 butterfly reductions (no barriers inside a wave)
// ---------------------------------------------------------------------------
__device__ __forceinline__ void wave_max_arg(float& v, int& i) {
    #pragma unroll
    for (int off = 16; off > 0; off >>= 1) {
        const float ov = __shfl_xor(v, off, 32);
        const int   oi = __shfl_xor(i, off, 32);
        if (ov > v || (ov == v && oi < i)) { v = ov; i = oi; }
    }
}
__device__ __forceinline__ float wave_sum(float v) {
    #pragma unroll
    for (int off = 16; off > 0; off >>= 1) v += __shfl_xor(v, off, 32);
    return v;
}

// ---------------------------------------------------------------------------
// Kernel B: sequential greedy decode, one block (WGP) per batch element.
// 512 threads = 16 wave32; thread n owns node n. K streams from L2 (16.8 MB,
// L2-resident). ctx gather uses the Tensor Data Mover in gather mode: two
// data-dependent rows of h[b] DMA'd into LDS, waited with s_wait_tensorcnt.
// ---------------------------------------------------------------------------
__global__ __launch_bounds__(512)
void decode_kernel(const float* __restrict__ h,
                   const float* __restrict__ Wp,
                   const float* __restrict__ Kall,
                   const float* __restrict__ qg,
                   const float* __restrict__ Wqn_t,
                   float* __restrict__ out_logp,   // [B][N][N]
                   float* __restrict__ out_seq) {  // [B][N]
    __shared__ float ctx[2 * DIM];
    __shared__ float q[DIM];
    __shared__ float wmaxv[NWAVE];
    __shared__ int   wmaxi[NWAVE];
    __shared__ float wsum[NWAVE];
    __shared__ unsigned char visited[NNODE];
    __shared__ int sel_first, sel_last;

    const int b    = blockIdx.x;
    const int n    = threadIdx.x;
    const int lane = n & 31;
    const int wv   = n >> 5;

    visited[n] = 0;
    if (n == 0) { sel_first = 0; sel_last = 0; }

    const float* Krow = Kall + ((size_t)b * NNODE + n) * DIM;
    __builtin_prefetch(Krow, 0, 1);     // global_prefetch_b8: warm this block's K slice
    __syncthreads();

    const unsigned long long hb_addr =
        (unsigned long long)(uintptr_t)(h + (size_t)b * NNODE * DIM);
    const unsigned int ctx_lds = (unsigned int)(uintptr_t)&ctx[0];

    for (int t = 0; t < NNODE; ++t) {
        // ---- context vector: [h_first ; h_last] (TDM row-gather), or Wp at t==0
        if (t == 0) {
            if (n < 2 * DIM) ctx[n] = Wp[n];
        } else {
#if USE_TDM
            if (wv == 0) {
                const int f = __builtin_amdgcn_readfirstlane(sel_first);
                const int l = __builtin_amdgcn_readfirstlane(sel_last);
                // D# group0: count=1, gather_mode=1, 32-bit indices, lds/global addr, type=2
                u32x4 g0;
                g0.x = 0xC0000001u;
                g0.y = ctx_lds;
                g0.z = (unsigned int)(hb_addr & 0xFFFFFFFFull);
                g0.w = (unsigned int)((hb_addr >> 32) & 0x01FFFFFFull) | 0x80000000u;
                // D# group1: data_size=4B, tensor_dim0=128, tensor_dim1=512,
                //            tile_dim0=128, tile_dim1=2 (two indices), dim0_stride=128
                i32x8 g1;
                g1[0] = 0x00020000;          // data_size=2 (4 bytes)
                g1[1] = (DIM << 16);         // tensor_dim0 = 128
                g1[2] = (NNODE << 16);       // tensor_dim1 = 512
                g1[3] = (DIM << 16);         // tile_dim0 = 128
                g1[4] = 2;                   // tile_dim1 = 2 valid indices
                g1[5] = DIM;                 // tensor_dim0_stride = 128
                g1[6] = 0;
                g1[7] = 0;
                // D# group2: 32-bit row indices {first, last}; group3/group4: zero
                i32x4 g2; g2[0] = f; g2[1] = l; g2[2] = 0; g2[3] = 0;
                i32x4 g3 = {};
                i32x8 g4 = {};   // 6-arg toolchain variant: extra descriptor group
                __builtin_amdgcn_tensor_load_to_lds(g0, g1, g2, g3, g4, 0);
                __builtin_amdgcn_s_wait_tensorcnt(0);
            }
#else
            if (n < 2 * DIM) {
                const int node = (n < DIM) ? sel_first : sel_last;
                ctx[n] = h[((size_t)b * NNODE + node) * DIM + (n & (DIM - 1))];
            }
#endif
        }
        __syncthreads();

        // ---- q = q_graph + ctx @ W_Q_nodes   (threads 0..127, contiguous reads)
        if (n < DIM) {
            float s = qg[(size_t)b * DIM + n];
            const float* w = Wqn_t + (size_t)n * (2 * DIM);
            #pragma unroll 8
            for (int d = 0; d < 2 * DIM; ++d) s += ctx[d] * w[d];
            q[n] = s;
        }
        __syncthreads();

        // ---- compat for node n: mean over heads of 10*tanh(scale * q.k)
        float cm = 0.f;
        #pragma unroll
        for (int hh = 0; hh < NHEAD; ++hh) {
            float dot = 0.f;
            #pragma unroll
            for (int k = 0; k < DK; ++k) dot += q[hh * DK + k] * Krow[hh * DK + k];
            cm += tanhf(SCALE * dot);
        }
        cm *= (10.0f / (float)NHEAD);

        const bool  vis = (visited[n] != 0);
        const float c   = vis ? -INFINITY : cm;

        // ---- max + argmax: wave butterfly, then 16-entry broadcast scan
        float v = c; int idx = n;
        wave_max_arg(v, idx);
        if (lane == 0) { wmaxv[wv] = v; wmaxi[wv] = idx; }
        __syncthreads();
        float mx = wmaxv[0]; int selected = wmaxi[0];
        #pragma unroll
        for (int w = 1; w < NWAVE; ++w) {
            const float o = wmaxv[w];
            if (o > mx || (o == mx && wmaxi[w] < selected)) { mx = o; selected = wmaxi[w]; }
        }

        // ---- sum of exp(c - mx): wave butterfly + broadcast scan
        float e = vis ? 0.f : __expf(c - mx);
        e = wave_sum(e);
        if (lane == 0) wsum[wv] = e;
        __syncthreads();
        float tot = 0.f;
        #pragma unroll
        for (int w = 0; w < NWAVE; ++w) tot += wsum[w];
        const float lse = __logf(tot) + mx;

        // ---- emit log-probs (streaming store: never re-read, keep K in L2)
        __builtin_nontemporal_store(vis ? -INFINITY : (c - lse),
                                    &out_logp[((size_t)b * NNODE + t) * NNODE + n]);

        // ---- greedy selection + decode state update
        if (n == selected) visited[n] = 1;
        if (n == 0) {
            out_seq[(size_t)b * NNODE + t] = (float)selected;
            sel_last = selected;
            if (t == 0) sel_first = selected;
        }
        __syncthreads();
    }
}

// ---------------------------------------------------------------------------
extern "C" void kernel_launch(void* const* d_in, const int* in_sizes, int n_in,
                              void* d_out, int out_size, void* d_ws, size_t ws_size,
                              hipStream_t stream) {
    const float* h   = (const float*)d_in[0];  // [64,512,128]
    const float* hg  = (const float*)d_in[1];  // [64,128]
    const float* Wqg = (const float*)d_in[2];  // [8,128,16]
    const float* Wqn = (const float*)d_in[3];  // [8,256,16]
    const float* Wk  = (const float*)d_in[4];  // [8,128,16]
    const float* Wp  = (const float*)d_in[5];  // [256]

    float* out_logp = (float*)d_out;                                   // [64,512,512]
    float* out_seq  = out_logp + (size_t)BATCH * NNODE * NNODE;        // [64,512]

    float* Kall  = (float*)d_ws;                                       // 16 MB
    float* qg    = Kall + (size_t)BATCH * NNODE * DIM;                 // 32 KB
    float* Wqn_t = qg + (size_t)BATCH * DIM;                           // 128 KB

    precompute_kernel<<<dim3(BATCH), dim3(256), 0, stream>>>(
        h, Wk, hg, Wqg, Wqn, Kall, qg, Wqn_t);
    decode_kernel<<<dim3(BATCH), dim3(NNODE), 0, stream>>>(
        h, Wp, Kall, qg, Wqn_t, out_logp, out_seq);
}